// Partition_enhanced_GIN_19078244729026
// MI455X (gfx1250) — compile-verified
//
#include <hip/hip_runtime.h>

// Problem constants (match reference)
#define NN   100000
#define EE   1600000
#define INC  96
#define HC   128
#define OUTC 128
#define GG   128

typedef float v2f __attribute__((ext_vector_type(2)));
typedef float v8f __attribute__((ext_vector_type(8)));

__device__ __forceinline__ v8f wmma_f32(v2f a, v2f b, v8f c) {
    // D(16x16,f32) = A(16x4,f32) x B(4x16,f32) + C
    return __builtin_amdgcn_wmma_f32_16x16x4_f32(
        /*neg_a=*/false, a, /*neg_b=*/false, b,
        /*c_mod=*/(short)0, c, /*reuse_a=*/false, /*reuse_b=*/false);
}

// x[N,128] = concat(node_feat[N,96], zeros)
__global__ void init_x_kernel(const float* __restrict__ nf, float* __restrict__ x) {
    int idx = blockIdx.x * blockDim.x + threadIdx.x;
    if (idx >= NN * HC) return;
    int r = idx >> 7, c = idx & 127;
    x[idx] = (c < INC) ? nf[r * INC + c] : 0.0f;
}

__global__ void zero_kernel(float4* __restrict__ p, int n4) {
    int idx = blockIdx.x * blockDim.x + threadIdx.x;
    if (idx < n4) p[idx] = make_float4(0.f, 0.f, 0.f, 0.f);
}

__global__ void zero_cnt_kernel(int* __restrict__ cnt) {
    if (threadIdx.x < 4) cnt[threadIdx.x] = 0;
}

// One pass over all edges: bucket edge ids by destination cluster (labels are static).
__global__ void build_lists_kernel(const int* __restrict__ ei, const int* __restrict__ labels,
                                   int* __restrict__ cnt, int* __restrict__ lists) {
    int e = blockIdx.x * blockDim.x + threadIdx.x;
    if (e >= EE) return;
    int dst = ei[EE + e];
    int l = labels[dst];
    int pos = atomicAdd(&cnt[l], 1);
    lists[(size_t)l * EE + pos] = e;
}

// Scatter-add over a pre-filtered edge list (only dst rows of the active cluster matter).
__global__ void scatter_filtered_kernel(const float* __restrict__ x, const int* __restrict__ ei,
                                        const int* __restrict__ list, const int* __restrict__ cnt_j,
                                        float* __restrict__ agg, int chunks) {
    long tid = (long)blockIdx.x * blockDim.x + threadIdx.x;
    int ch = (int)(tid & 31);
    int ii = (int)(tid >> 5);
    if (ii >= *cnt_j) return;            // whole wave shares ii range -> coherent exit
    if (ch >= chunks) return;
    int e   = list[ii];
    int src = ei[e];
    int dst = ei[EE + e];
    const float4 v = *(const float4*)(x + (size_t)src * HC + ch * 4);
    float* d = agg + (size_t)dst * HC + ch * 4;
    atomicAdd(d + 0, v.x);
    atomicAdd(d + 1, v.y);
    atomicAdd(d + 2, v.z);
    atomicAdd(d + 3, v.w);
}

// Fallback: unfiltered scatter with inline dst-cluster check (used if ws too small for lists).
__global__ void scatter_kernel(const float* __restrict__ x, const int* __restrict__ ei,
                               const int* __restrict__ labels, float* __restrict__ agg,
                               int jc, int chunks) {
    long tid = (long)blockIdx.x * blockDim.x + threadIdx.x;
    if (tid >= (long)EE * 32) return;
    int ch = (int)(tid & 31);
    if (ch >= chunks) return;
    int e  = (int)(tid >> 5);
    int dst = ei[EE + e];
    if (labels[dst] != jc) return;
    int src = ei[e];
    const float4 v = *(const float4*)(x + (size_t)src * HC + ch * 4);
    float* d = agg + (size_t)dst * HC + ch * 4;
    atomicAdd(d + 0, v.x);
    atomicAdd(d + 1, v.y);
    atomicAdd(d + 2, v.z);
    atomicAdd(d + 3, v.w);
}

// Per-node 2-layer MLP with fp32 WMMA. One wave (wave32) owns a 16-row tile; 4 waves/block.
// k-outer / n-inner: each A fragment loads once and feeds 8 independent WMMAs (8 live v8f accs).
// Branchless A prep (clamped row + 0/1 scale factors) keeps EXEC untouched in the hot loop.
__global__ void __launch_bounds__(128)
mlp_nodes_kernel(float* __restrict__ x, const float* __restrict__ agg,
                 const int* __restrict__ labels,
                 const float* __restrict__ W1, const float* __restrict__ b1,
                 const float* __restrict__ W2, const float* __restrict__ b2,
                 int jc, int Kin) {
    extern __shared__ float t_s[];                 // 64 x 128 floats (32 KB)
    __builtin_prefetch(W1, 0, 1);
    __builtin_prefetch(W2, 0, 1);
    const int w     = threadIdx.x >> 5;
    const int lane  = threadIdx.x & 31;
    const int l15   = lane & 15;
    const int kofs  = (lane >> 4) << 1;            // lanes 0-15: K+0/1; 16-31: K+2/3
    const int rowbase = blockIdx.x * 64 + w * 16;
    const int myrow = rowbase + l15;               // A-matrix row owned by this lane
    const bool inr  = myrow < NN;
    const int  rowsafe = inr ? myrow : 0;          // clamp: always-legal address
    const bool m    = inr && (labels[rowsafe] == jc);
    const float s   = inr ? 1.0f : 0.0f;
    const float mm  = m   ? 1.0f : 0.0f;
    const float* aggrow = agg + (size_t)rowsafe * HC;
    const float* xrow   = x   + (size_t)rowsafe * HC;
    const int ks1 = Kin >> 2;

    const v8f vzero = {0.f,0.f,0.f,0.f,0.f,0.f,0.f,0.f};
    v8f acc[8];
#pragma unroll
    for (int n = 0; n < 8; ++n) acc[n] = vzero;

    // stage 1: t = relu((agg + mask*xin) @ W1 + b1)
    for (int kk = 0; kk < ks1; ++kk) {
        const int c = kk * 4 + kofs;               // even -> 8B-aligned b64 loads
        const v2f av = *(const v2f*)(aggrow + c);
        const v2f xv = *(const v2f*)(xrow + c);
        v2f a;
        a.x = s * (av.x + mm * xv.x);              // (1+eps)*xin residual, eps=0
        a.y = s * (av.y + mm * xv.y);
        const float* Wc = W1 + c * HC + l15;
#pragma unroll
        for (int n = 0; n < 8; ++n) {
            v2f b;
            b.x = Wc[n * 16];
            b.y = Wc[HC + n * 16];
            acc[n] = wmma_f32(a, b, acc[n]);
        }
    }
#pragma unroll
    for (int n = 0; n < 8; ++n) {
        const int ncol = n * 16 + l15;
        const float bias = b1[ncol];
#pragma unroll
        for (int g = 0; g < 8; ++g) {
            const int mrow = (lane < 16) ? g : (g + 8);        // D layout: M=g / M=g+8
            t_s[(w * 16 + mrow) * HC + ncol] = fmaxf(acc[n][g] + bias, 0.f);
        }
    }

    // per-lane write masks for the 8 output rows this lane's D-VGPRs cover
    bool wm[8];
#pragma unroll
    for (int g = 0; g < 8; ++g) {
        const int mrow = (lane < 16) ? g : (g + 8);
        const int grow = rowbase + mrow;
        wm[g] = (grow < NN) && (labels[grow < NN ? grow : 0] == jc);
    }

    // stage 2: h = t @ W2 + b2 ; masked write-back into x
#pragma unroll
    for (int n = 0; n < 8; ++n) acc[n] = vzero;
    const float* trow = t_s + (w * 16 + l15) * HC;
    for (int kk = 0; kk < 32; ++kk) {
        const int c = kk * 4 + kofs;
        const v2f a = *(const v2f*)(trow + c);     // ds_load_b64
        const float* Wc = W2 + c * HC + l15;
#pragma unroll
        for (int n = 0; n < 8; ++n) {
            v2f b;
            b.x = Wc[n * 16];
            b.y = Wc[HC + n * 16];
            acc[n] = wmma_f32(a, b, acc[n]);
        }
    }
#pragma unroll
    for (int n = 0; n < 8; ++n) {
        const int ncol = n * 16 + l15;
        const float bias = b2[ncol];
#pragma unroll
        for (int g = 0; g < 8; ++g) {
            if (wm[g]) {
                const int mrow = (lane < 16) ? g : (g + 8);
                x[(size_t)(rowbase + mrow) * HC + ncol] = acc[n][g] + bias;
            }
        }
    }
}

// global_add_pool: pooled[batch[r]] += x[r]
__global__ void pool_scatter_kernel(const float* __restrict__ x, const int* __restrict__ batch,
                                    float* __restrict__ pooled) {
    long tid = (long)blockIdx.x * blockDim.x + threadIdx.x;
    if (tid >= (long)NN * 32) return;
    int ch = (int)(tid & 31);
    int r  = (int)(tid >> 5);
    int g  = batch[r];
    const float4 v = *(const float4*)(x + (size_t)r * HC + ch * 4);
    float* d = pooled + (size_t)g * HC + ch * 4;
    atomicAdd(d + 0, v.x);
    atomicAdd(d + 1, v.y);
    atomicAdd(d + 2, v.z);
    atomicAdd(d + 3, v.w);
}

// Final pooled MLP: [128,128] -> relu -> [128,128]. 2 blocks x 64 rows.
__global__ void __launch_bounds__(128)
mlp_dense_kernel(const float* __restrict__ in,
                 const float* __restrict__ W1, const float* __restrict__ b1,
                 const float* __restrict__ W2, const float* __restrict__ b2,
                 float* __restrict__ out) {
    extern __shared__ float t_s[];                 // 64 x 128 floats
    __builtin_prefetch(W1, 0, 1);
    __builtin_prefetch(W2, 0, 1);
    const int w    = threadIdx.x >> 5;
    const int lane = threadIdx.x & 31;
    const int l15  = lane & 15;
    const int kofs = (lane >> 4) << 1;
    const int rowbase = blockIdx.x * 64 + w * 16;
    const float* inrow = in + (size_t)(rowbase + l15) * HC;

    const v8f vzero = {0.f,0.f,0.f,0.f,0.f,0.f,0.f,0.f};
    v8f acc[8];
#pragma unroll
    for (int n = 0; n < 8; ++n) acc[n] = vzero;

    for (int kk = 0; kk < 32; ++kk) {
        const int c = kk * 4 + kofs;
        const v2f a = *(const v2f*)(inrow + c);
        const float* Wc = W1 + c * HC + l15;
#pragma unroll
        for (int n = 0; n < 8; ++n) {
            v2f b;
            b.x = Wc[n * 16];
            b.y = Wc[HC + n * 16];
            acc[n] = wmma_f32(a, b, acc[n]);
        }
    }
#pragma unroll
    for (int n = 0; n < 8; ++n) {
        const int ncol = n * 16 + l15;
        const float bias = b1[ncol];
#pragma unroll
        for (int g = 0; g < 8; ++g) {
            const int mrow = (lane < 16) ? g : (g + 8);
            t_s[(w * 16 + mrow) * HC + ncol] = fmaxf(acc[n][g] + bias, 0.f);
        }
    }

#pragma unroll
    for (int n = 0; n < 8; ++n) acc[n] = vzero;
    const float* trow = t_s + (w * 16 + l15) * HC;
    for (int kk = 0; kk < 32; ++kk) {
        const int c = kk * 4 + kofs;
        const v2f a = *(const v2f*)(trow + c);
        const float* Wc = W2 + c * HC + l15;
#pragma unroll
        for (int n = 0; n < 8; ++n) {
            v2f b;
            b.x = Wc[n * 16];
            b.y = Wc[HC + n * 16];
            acc[n] = wmma_f32(a, b, acc[n]);
        }
    }
#pragma unroll
    for (int n = 0; n < 8; ++n) {
        const int ncol = n * 16 + l15;
        const float bias = b2[ncol];
#pragma unroll
        for (int g = 0; g < 8; ++g) {
            const int mrow = (lane < 16) ? g : (g + 8);
            out[(size_t)(rowbase + mrow) * OUTC + ncol] = acc[n][g] + bias;
        }
    }
}

extern "C" void kernel_launch(void* const* d_in, const int* in_sizes, int n_in,
                              void* d_out, int out_size, void* d_ws, size_t ws_size,
                              hipStream_t stream) {
    const float* node_feat = (const float*)d_in[0];
    const int*   labels    = (const int*)d_in[1];
    const int*   ei        = (const int*)d_in[2];
    const int*   batch     = (const int*)d_in[3];
    const float* c0W1 = (const float*)d_in[4];
    const float* c0b1 = (const float*)d_in[5];
    const float* c0W2 = (const float*)d_in[6];
    const float* c0b2 = (const float*)d_in[7];
    const float* cRW1 = (const float*)d_in[8];
    const float* cRb1 = (const float*)d_in[9];
    const float* cRW2 = (const float*)d_in[10];
    const float* cRb2 = (const float*)d_in[11];
    const float* pW1  = (const float*)d_in[12];
    const float* pb1  = (const float*)d_in[13];
    const float* pW2  = (const float*)d_in[14];
    const float* pb2  = (const float*)d_in[15];
    float* out = (float*)d_out;

    // workspace: x[N,128] | agg[N,128] | pooled[G,128] | cnt[4] | lists[4*E]
    float* x      = (float*)d_ws;
    float* agg    = x + (size_t)NN * HC;
    float* pooled = agg + (size_t)NN * HC;
    int*   cnt    = (int*)(pooled + (size_t)GG * HC);
    int*   lists  = cnt + 4;
    const size_t need_filtered =
        ((size_t)2 * NN * HC + (size_t)GG * HC) * sizeof(float) + (4 + (size_t)4 * EE) * sizeof(int);
    const bool use_filtered = ws_size >= need_filtered;

    init_x_kernel<<<(NN * HC + 255) / 256, 256, 0, stream>>>(node_feat, x);

    if (use_filtered) {
        zero_cnt_kernel<<<1, 32, 0, stream>>>(cnt);
        build_lists_kernel<<<(EE + 255) / 256, 256, 0, stream>>>(ei, labels, cnt, lists);
    }

    const long sc_total  = (long)EE * 32;
    const int  sc_blocks = (int)((sc_total + 255) / 256);
    const int  mlp_blocks = (NN + 63) / 64;
    const size_t smem = (size_t)64 * HC * sizeof(float);

    for (int i = 0; i < 3; ++i) {
        const int Kin = (i == 0) ? INC : HC;
        for (int jc = 0; jc < 4; ++jc) {
            zero_kernel<<<(NN * HC / 4 + 255) / 256, 256, 0, stream>>>((float4*)agg, NN * HC / 4);
            if (use_filtered) {
                scatter_filtered_kernel<<<sc_blocks, 256, 0, stream>>>(
                    x, ei, lists + (size_t)jc * EE, cnt + jc, agg, Kin / 4);
            } else {
                scatter_kernel<<<sc_blocks, 256, 0, stream>>>(x, ei, labels, agg, jc, Kin / 4);
            }
            const float *W1, *b1, *W2, *b2;
            if (i == 0) {
                W1 = c0W1 + (size_t)jc * INC * HC;
                b1 = c0b1 + jc * HC;
                W2 = c0W2 + (size_t)jc * HC * HC;
                b2 = c0b2 + jc * HC;
            } else {
                const int k = (i - 1) * 4 + jc;
                W1 = cRW1 + (size_t)k * HC * HC;
                b1 = cRb1 + k * HC;
                W2 = cRW2 + (size_t)k * HC * HC;
                b2 = cRb2 + k * HC;
            }
            mlp_nodes_kernel<<<mlp_blocks, 128, smem, stream>>>(
                x, agg, labels, W1, b1, W2, b2, jc, Kin);
        }
    }

    zero_kernel<<<(GG * HC / 4 + 255) / 256, 256, 0, stream>>>((float4*)pooled, GG * HC / 4);
    pool_scatter_kernel<<<(int)(((long)NN * 32 + 255) / 256), 256, 0, stream>>>(x, batch, pooled);
    mlp_dense_kernel<<<GG / 64, 128, smem, stream>>>(pooled, pW1, pb1, pW2, pb2, out);
}